// HungarianMatcher_56384330661983
// MI455X (gfx1250) — compile-verified
//
#include <hip/hip_runtime.h>
#include <hip/hip_bf16.h>
#include <math.h>

// Problem constants (from reference): B=16, Q=900, C=4, T=100
#define NB 16
#define NQ 900
#define NC 4
#define NT 100
#define BQ (NB * NQ)          // 14400
#define BT (NB * NT)          // 1600
#define LSA_INF 1e18

typedef float v2f __attribute__((ext_vector_type(2)));
typedef float v8f __attribute__((ext_vector_type(8)));

// Shared cost expression: identical FP ops in kernel 2 and kernel 3 so the
// LSA input is bit-identical to the reported C matrix.
// cost = 5 * sum_c |p_c - t_c| - p[label]   (class term passed in as cls)
__device__ __forceinline__ float hm_cost(const float4 pm, const float4 tb,
                                         const float cls) {
  float bb = fabsf(pm.x - tb.x);
  bb = __fadd_rn(bb, fabsf(pm.y - tb.y));
  bb = __fadd_rn(bb, fabsf(pm.z - tb.z));
  bb = __fadd_rn(bb, fabsf(pm.w - tb.w));
  return fmaf(5.0f, bb, -cls);   // explicit FMA: deterministic in both users
}

// ---------------------------------------------------------------------------
// Kernel 1: softmax over the 4 logits per (b,q) row -> prob[BQ][4] in ws
// ---------------------------------------------------------------------------
__global__ void hm_softmax_kernel(const float* __restrict__ logits,
                                  float* __restrict__ prob) {
  int idx = blockIdx.x * blockDim.x + threadIdx.x;
  if (idx >= BQ) return;
  float4 x = reinterpret_cast<const float4*>(logits)[idx];
  float mx = fmaxf(fmaxf(x.x, x.y), fmaxf(x.z, x.w));
  float e0 = expf(x.x - mx), e1 = expf(x.y - mx);
  float e2 = expf(x.z - mx), e3 = expf(x.w - mx);
  float s = ((e0 + e1) + e2) + e3;
  float inv = 1.0f / s;
  float4 p;
  p.x = e0 * inv; p.y = e1 * inv; p.z = e2 * inv; p.w = e3 * inv;
  reinterpret_cast<float4*>(prob)[idx] = p;
}

// ---------------------------------------------------------------------------
// Kernel 2: full cost matrix C[BQ][BT] via WMMA f32 16x16x4.
// Class term as GEMM prob[16x4] x onehot(labels)[4x16]; L1-box term in VALU.
// One wave per 16x16 tile; 900 x 100 tiles; blockDim=256 (8 waves).
// ---------------------------------------------------------------------------
__global__ __launch_bounds__(256)
void hm_cost_kernel(const float* __restrict__ prob,
                    const int* __restrict__ labels,
                    const float* __restrict__ tboxes,
                    float* __restrict__ Cout) {
  const int waveInBlk = threadIdx.x >> 5;
  const int lane = threadIdx.x & 31;
  const int w = blockIdx.x * 8 + waveInBlk;     // 0..89999
  const int tileM = w / 100;                    // 0..899  (bq tiles)
  const int tileN = w - tileM * 100;            // 0..99   (target tiles)
  const int bq0 = tileM * 16;
  const int j0 = tileN * 16;
  const int lh = lane & 15;
  const bool hi = lane >= 16;

  // A matrix (16x4 f32): lanes 0-15 -> {K0,K1}, lanes 16-31 -> {K2,K3}
  const float4 pr = reinterpret_cast<const float4*>(prob)[bq0 + lh];
  v2f a;
  a.x = hi ? pr.z : pr.x;
  a.y = hi ? pr.w : pr.y;

  // B matrix (4x16 f32) one-hot of labels: lanes 0-15 -> rows {K0,K1},
  // lanes 16-31 -> rows {K2,K3}; column = lane&15
  const int lab = labels[j0 + lh];
  const int kb = hi ? 2 : 0;
  v2f bm;
  bm.x = (lab == kb) ? 1.0f : 0.0f;
  bm.y = (lab == kb + 1) ? 1.0f : 0.0f;

  v8f cacc = {};
  // D[m][n] = prob[bq0+m][ labels[j0+n] ]  (exact: one p*1 plus exact zeros)
  v8f d = __builtin_amdgcn_wmma_f32_16x16x4_f32(
      /*neg_a=*/false, a, /*neg_b=*/false, bm,
      /*c_mod=*/(short)0, cacc, /*reuse_a=*/false, /*reuse_b=*/false);

  // C/D layout: VGPR k -> (M = k + (hi?8:0), N = lane&15)
  const float4 tb = reinterpret_cast<const float4*>(tboxes)[j0 + lh];
  const int mbase = bq0 + (hi ? 8 : 0);
#pragma unroll
  for (int k = 0; k < 8; ++k) {
    const float4 pm = reinterpret_cast<const float4*>(prob)[mbase + k];
    const float cost = hm_cost(pm, tb, d[k]);
    __builtin_nontemporal_store(
        cost, Cout + (size_t)(mbase + k) * BT + (size_t)(j0 + lh));
  }
}

// ---------------------------------------------------------------------------
// Kernel 3: per-batch transposed cost ct[b][i][j] = C[b][j][b*100+i] (f32),
// recomputed from inputs with identical FP ops as kernel 2 (hm_cost; the
// WMMA class term equals p[label] exactly, so cls matches bitwise).
// ---------------------------------------------------------------------------
__global__ void hm_ctrans_kernel(const float* __restrict__ prob,
                                 const int* __restrict__ labels,
                                 const float* __restrict__ tboxes,
                                 float* __restrict__ ct) {
  int t = blockIdx.x * blockDim.x + threadIdx.x;  // 16*100*900 = 1,440,000
  if (t >= NB * NT * NQ) return;
  int j = t % NQ;
  int r = t / NQ;
  int i = r % NT;
  int b = r / NT;
  const float4 pm = reinterpret_cast<const float4*>(prob)[b * NQ + j];
  int gt = b * NT + i;
  const float4 tb = reinterpret_cast<const float4*>(tboxes)[gt];
  int lab = labels[gt];
  float cls = (lab == 0) ? pm.x : (lab == 1) ? pm.y : (lab == 2) ? pm.z : pm.w;
  ct[t] = hm_cost(pm, tb, cls);
}

// ---------------------------------------------------------------------------
// Kernel 4: Jonker-Volgenant LSA (float64), one wave32 per batch.
// Solves the transposed problem (n=100 target rows, m=900 query cols),
// exactly mirroring the reference _lsa including first-index argmin.
// ---------------------------------------------------------------------------
__global__ __launch_bounds__(32)
void hm_lsa_kernel(const float* __restrict__ ct,
                   float* __restrict__ rowOut,
                   float* __restrict__ colOut) {
  const int b = blockIdx.x;
  const int tid = threadIdx.x;
  const int n = NT, m = NQ;
  const float* cb = ct + (size_t)b * n * m;

  __shared__ double v_[NQ + 1];
  __shared__ double u_[NT + 1];
  __shared__ double minv_[NQ + 1];
  __shared__ int p_[NQ + 1];
  __shared__ int way_[NQ + 1];
  __shared__ unsigned char used_[NQ + 1];
  __shared__ double deltaS;
  __shared__ int j1S;

  for (int j = tid; j <= m; j += 32) { v_[j] = 0.0; p_[j] = 0; }
  for (int i = tid; i <= n; i += 32) u_[i] = 0.0;
  __syncthreads();

  for (int i = 1; i <= n; ++i) {
    if (tid == 0) p_[0] = i;
    for (int j = tid; j <= m; j += 32) { minv_[j] = LSA_INF; used_[j] = 0; }
    __syncthreads();
    int j0 = 0;
    while (true) {
      if (tid == 0) used_[j0] = 1;
      __syncthreads();
      const int i0 = p_[j0];
      const double ui0 = u_[i0];
      const float* crow = cb + (size_t)(i0 - 1) * m;

      // relaxation + local first-min over this thread's columns
      double bv = LSA_INF;
      int bj = m + 1;
      for (int j = tid + 1; j <= m; j += 32) {
        if (!used_[j]) {
          double cur = (double)crow[j - 1] - ui0 - v_[j];
          if (cur < minv_[j]) { minv_[j] = cur; way_[j] = j0; }
          double mv = minv_[j];
          if (mv < bv) { bv = mv; bj = j; }
        }
      }
      // wave-wide min with first-index tie-break
      for (int off = 16; off > 0; off >>= 1) {
        double ov = __shfl_down(bv, off, 32);
        int oj = __shfl_down(bj, off, 32);
        if (ov < bv || (ov == bv && oj < bj)) { bv = ov; bj = oj; }
      }
      if (tid == 0) { deltaS = bv; j1S = bj; }
      __syncthreads();
      const double delta = deltaS;
      const int j1 = j1S;
      for (int j = tid; j <= m; j += 32) {
        if (used_[j]) {
          u_[p_[j]] += delta;  // distinct rows per used column -> race-free
          v_[j] -= delta;
        } else {
          minv_[j] -= delta;
        }
      }
      __syncthreads();
      j0 = j1;
      if (p_[j0] == 0) break;
    }
    if (tid == 0) {
      int jj = j0;
      while (jj != 0) { int jp = way_[jj]; p_[jj] = p_[jp]; jj = jp; }
    }
    __syncthreads();
  }

  // Emit pairs sorted by query index (ascending j == reference stable order)
  if (tid == 0) {
    int k = 0;
    for (int j = 1; j <= m; ++j) {
      if (p_[j] > 0) {
        rowOut[b * NT + k] = (float)(j - 1);       // query index
        colOut[b * NT + k] = (float)(p_[j] - 1);   // target index
        ++k;
      }
    }
  }
}

// ---------------------------------------------------------------------------
// Launcher
// Inputs: d_in[0]=pred_logits f32 [16,900,4], d_in[1]=pred_boxes (unused),
//         d_in[2]=tgt_labels int [16,100], d_in[3]=tgt_boxes f32 [16,100,4]
// Output d_out (f32): row_idx[1600] | col_idx[1600] | C[16*900*1600]
// ---------------------------------------------------------------------------
extern "C" void kernel_launch(void* const* d_in, const int* in_sizes, int n_in,
                              void* d_out, int out_size, void* d_ws, size_t ws_size,
                              hipStream_t stream) {
  (void)in_sizes; (void)n_in; (void)out_size; (void)ws_size;
  const float* logits = (const float*)d_in[0];
  const int* labels   = (const int*)d_in[2];
  const float* tboxes = (const float*)d_in[3];

  float* out = (float*)d_out;
  float* rowOut = out;              // [16*100]
  float* colOut = out + BT;         // [16*100]
  float* Cout   = out + 2 * BT;     // [14400*1600]

  char* ws = (char*)d_ws;
  float* prob = (float*)ws;                         // BQ*4 floats   (230,400 B)
  float* ct   = (float*)(ws + (size_t)BQ * 4 * 4);  // 16*100*900 f  (5,760,000 B)

  // 1) softmax
  hm_softmax_kernel<<<(BQ + 255) / 256, 256, 0, stream>>>(logits, prob);

  // 2) cost matrix via WMMA: 900x100 tiles of 16x16, 8 waves/block
  hm_cost_kernel<<<(900 * 100) / 8, 256, 0, stream>>>(prob, labels, tboxes, Cout);

  // 3) compact per-batch transposed cost (bit-identical to Cout values)
  hm_ctrans_kernel<<<(NB * NT * NQ) / 256, 256, 0, stream>>>(prob, labels, tboxes, ct);

  // 4) Hungarian per batch (wave32 JV)
  hm_lsa_kernel<<<NB, 32, 0, stream>>>(ct, rowOut, colOut);
}